// GNNRegressor_50861002719555
// MI455X (gfx1250) — compile-verified
//
#include <hip/hip_runtime.h>

#define N_NODES 100000
#define N_EDGES 1000000
#define N_GRAPH 2000
#define ND 64
#define ED 16
#define HD 64
#define NLAYERS 4

#define EDGE_WAVES 4
#define MLP_WAVES 4

typedef __attribute__((ext_vector_type(2))) float v2f;
typedef __attribute__((ext_vector_type(8))) float v8f;

// D = A(16x4 f32) * B(4x16 f32) + C(16x16 f32), full fp32 WMMA (CDNA5)
__device__ __forceinline__ v8f wmma_f32_16x16x4(v2f a, v2f b, v8f c) {
  return __builtin_amdgcn_wmma_f32_16x16x4_f32(false, a, false, b, (short)0, c,
                                               false, false);
}

// ---------------------------------------------------------------------------
// h[i] = (1 + eps[l]) * x[i]   (seed for the scatter-add aggregation)
// ---------------------------------------------------------------------------
__global__ void init_h_kernel(const float* __restrict__ xin,
                              const float* __restrict__ eps, int layer,
                              float* __restrict__ h) {
  int i = blockIdx.x * blockDim.x + threadIdx.x;
  if (i < N_NODES * ND) {
    float s = 1.0f + eps[layer];
    h[i] = s * xin[i];
  }
}

// ---------------------------------------------------------------------------
// Fused GINE edge stage: e = edge_attr @ lin_e[l]  (WMMA f32 16x16x4, K=16)
//                        m = relu(x[src] + e); atomicAdd(h[dst], m)
// One wave handles a 16-edge tile; 4 column-tiles of 16 over ND=64.
// ---------------------------------------------------------------------------
__global__ void __launch_bounds__(32 * EDGE_WAVES)
edge_kernel(const float* __restrict__ x, const int* __restrict__ ei,
            const float* __restrict__ ea, const float* __restrict__ linE,
            float* __restrict__ h) {
  const int lane = threadIdx.x & 31;
  const int wave = threadIdx.x >> 5;
  const int n = lane & 15;   // column within 16-wide tile / A row id
  const int hi = lane >> 4;  // lane-half selector

  // Preload B fragments of lin_e [16 x 64] once per wave.
  // B layout (4x16 f32): v0 holds K = kb+2*hi, v1 holds K = kb+2*hi+1, N = n.
  v2f bf[4][4];
#pragma unroll
  for (int nt = 0; nt < 4; ++nt)
#pragma unroll
    for (int kk = 0; kk < 4; ++kk) {
      int k0 = 4 * kk + 2 * hi;
      v2f t = {linE[k0 * ND + 16 * nt + n], linE[(k0 + 1) * ND + 16 * nt + n]};
      bf[nt][kk] = t;
    }

  const int nTiles = N_EDGES / 16;  // 62500 exactly
  for (int tile = blockIdx.x * EDGE_WAVES + wave; tile < nTiles;
       tile += gridDim.x * EDGE_WAVES) {
    int base = tile * 16;

    // This lane owns D-rows M = 8*hi + v (v = 0..7): load their src/dst
    // indices directly as two int4 (b128) loads per array. All 16 lanes of a
    // half read the same addresses -> L0 broadcast, no cross-lane permutes.
    int4 s0 = *(const int4*)(ei + base + 8 * hi);
    int4 s1 = *(const int4*)(ei + base + 8 * hi + 4);
    int4 d0 = *(const int4*)(ei + N_EDGES + base + 8 * hi);
    int4 d1 = *(const int4*)(ei + N_EDGES + base + 8 * hi + 4);
    // Unsigned flat element offsets (row*64 + n): zext-foldable into the
    // SADDR + 32-bit-voffset (scale_offset) addressing form for both the
    // gathers and the atomics.
    unsigned sOff[8], dOff[8];
    {
      int sIdx[8] = {s0.x, s0.y, s0.z, s0.w, s1.x, s1.y, s1.z, s1.w};
      int dIdx[8] = {d0.x, d0.y, d0.z, d0.w, d1.x, d1.y, d1.z, d1.w};
#pragma unroll
      for (int v = 0; v < 8; ++v) {
        sOff[v] = (unsigned)(sIdx[v] * ND + n);
        dOff[v] = (unsigned)(dIdx[v] * ND + n);
      }
    }

    // A fragments: row M = n of edge_attr, cols 4*kk + 2*hi + {0,1}
    v2f af[4];
#pragma unroll
    for (int kk = 0; kk < 4; ++kk)
      af[kk] = *(const v2f*)(ea + (base + n) * ED + 4 * kk + 2 * hi);

#pragma unroll
    for (int nt = 0; nt < 4; ++nt) {
      v8f acc = {};
#pragma unroll
      for (int kk = 0; kk < 4; ++kk) acc = wmma_f32_16x16x4(af[kk], bf[nt][kk], acc);
      // Batch the 8 gathers so they clause and overlap, then drain into
      // ReLU + atomic scatter.
      float xv[8];
#pragma unroll
      for (int v = 0; v < 8; ++v) xv[v] = x[sOff[v] + 16u * nt];
#pragma unroll
      for (int v = 0; v < 8; ++v) {
        float m = fmaxf(xv[v] + acc[v], 0.0f);
        atomicAdd(h + dOff[v] + 16u * nt, m);
      }
    }
  }
}

// ---------------------------------------------------------------------------
// Node MLP: x_out = relu( relu(h @ W1 + b1) @ W2 + b2 )
// One wave per 16-node tile; W1/W2 staged in LDS; intermediate tile staged
// through per-wave LDS scratch to convert WMMA D-layout -> A-layout.
// ---------------------------------------------------------------------------
__global__ void __launch_bounds__(32 * MLP_WAVES)
node_mlp_kernel(const float* __restrict__ h, const float* __restrict__ W1,
                const float* __restrict__ b1, const float* __restrict__ W2,
                const float* __restrict__ b2, float* __restrict__ xout,
                int nTiles) {
  __shared__ float sW1[HD * HD];
  __shared__ float sW2[HD * HD];
  __shared__ __align__(16) float sT[MLP_WAVES][16 * HD];

  for (int i = threadIdx.x; i < HD * HD; i += blockDim.x) {
    sW1[i] = W1[i];
    sW2[i] = W2[i];
  }
  __syncthreads();

  const int lane = threadIdx.x & 31;
  const int wave = threadIdx.x >> 5;
  const int n = lane & 15;
  const int hi = lane >> 4;

  int tile = blockIdx.x * MLP_WAVES + wave;
  if (tile >= nTiles) return;  // wave-uniform; no barriers beyond this point
  int rowBase = tile * 16;
  float* tptr = &sT[wave][0];

  // ---- GEMM1: t = relu(h_tile @ W1 + b1) ----
  v2f af[16];
#pragma unroll
  for (int kk = 0; kk < 16; ++kk)
    af[kk] = *(const v2f*)(h + (unsigned)((rowBase + n) * HD + 4 * kk + 2 * hi));

#pragma unroll
  for (int nt = 0; nt < 4; ++nt) {
    v8f acc = {};
#pragma unroll
    for (int kk = 0; kk < 16; ++kk) {
      int k0 = 4 * kk + 2 * hi;
      v2f bfr = {sW1[k0 * HD + 16 * nt + n], sW1[(k0 + 1) * HD + 16 * nt + n]};
      acc = wmma_f32_16x16x4(af[kk], bfr, acc);
    }
    float bias = b1[16 * nt + n];
#pragma unroll
    for (int v = 0; v < 8; ++v) {
      float t = fmaxf(acc[v] + bias, 0.0f);
      tptr[(8 * hi + v) * HD + 16 * nt + n] = t;  // D-layout -> LDS row-major
    }
  }
  // Ensure LDS stores completed before re-reading in A-layout (CDNA5 counter)
  asm volatile("s_wait_dscnt 0" ::: "memory");

  // ---- GEMM2: x_out = relu(t @ W2 + b2) ----
#pragma unroll
  for (int kk = 0; kk < 16; ++kk)
    af[kk] = *(const v2f*)(tptr + n * HD + 4 * kk + 2 * hi);

#pragma unroll
  for (int nt = 0; nt < 4; ++nt) {
    v8f acc = {};
#pragma unroll
    for (int kk = 0; kk < 16; ++kk) {
      int k0 = 4 * kk + 2 * hi;
      v2f bfr = {sW2[k0 * HD + 16 * nt + n], sW2[(k0 + 1) * HD + 16 * nt + n]};
      acc = wmma_f32_16x16x4(af[kk], bfr, acc);
    }
    float bias = b2[16 * nt + n];
#pragma unroll
    for (int v = 0; v < 8; ++v) {
      float t = fmaxf(acc[v] + bias, 0.0f);
      xout[(unsigned)((rowBase + 8 * hi + v) * HD + 16 * nt + n)] = t;
    }
  }
}

// ---------------------------------------------------------------------------
// Pooling + head
// ---------------------------------------------------------------------------
__global__ void zero_pool_kernel(float* __restrict__ sums, float* __restrict__ cnt) {
  int i = blockIdx.x * blockDim.x + threadIdx.x;
  if (i < N_GRAPH * HD) sums[i] = 0.0f;
  if (i < N_GRAPH) cnt[i] = 0.0f;
}

__global__ void pool_kernel(const float* __restrict__ x,
                            const int* __restrict__ batch,
                            float* __restrict__ sums, float* __restrict__ cnt) {
  int i = blockIdx.x * blockDim.x + threadIdx.x;
  if (i >= N_NODES * HD) return;
  int node = i >> 6;
  int col = i & 63;
  int b = batch[node];
  atomicAdd(sums + (unsigned)(b * HD + col), x[i]);
  if (col == 0) atomicAdd(cnt + b, 1.0f);
}

__global__ void head_kernel(const float* __restrict__ sums,
                            const float* __restrict__ cnt,
                            const float* __restrict__ tcond,
                            const float* __restrict__ hW1,
                            const float* __restrict__ hb1,
                            const float* __restrict__ hW2,
                            const float* __restrict__ hb2,
                            float* __restrict__ out) {
  int g = blockIdx.x * blockDim.x + threadIdx.x;
  if (g >= N_GRAPH) return;
  float inv = 1.0f / fmaxf(cnt[g], 1.0f);
  float p[HD];
#pragma unroll
  for (int c = 0; c < HD; ++c) p[c] = sums[g * HD + c] * inv;
  float t = tcond[g];
  float acc = hb2[0];
  for (int j = 0; j < HD; ++j) {
    float s = hb1[j];
#pragma unroll
    for (int c = 0; c < HD; ++c) s += p[c] * hW1[c * HD + j];
    s += t * hW1[HD * HD + j];  // t_cond row of head_W1 (row 64)
    acc += fmaxf(s, 0.0f) * hW2[j];
  }
  out[g] = acc;
}

// ---------------------------------------------------------------------------
extern "C" void kernel_launch(void* const* d_in, const int* in_sizes, int n_in,
                              void* d_out, int out_size, void* d_ws,
                              size_t ws_size, hipStream_t stream) {
  const float* x0    = (const float*)d_in[0];
  const int*   ei    = (const int*)d_in[1];
  const float* ea    = (const float*)d_in[2];
  const int*   batch = (const int*)d_in[3];
  const float* tc    = (const float*)d_in[4];
  const float* linE  = (const float*)d_in[5];
  const float* W1    = (const float*)d_in[6];
  const float* b1    = (const float*)d_in[7];
  const float* W2    = (const float*)d_in[8];
  const float* b2    = (const float*)d_in[9];
  const float* eps   = (const float*)d_in[10];
  const float* hW1   = (const float*)d_in[11];
  const float* hb1   = (const float*)d_in[12];
  const float* hW2   = (const float*)d_in[13];
  const float* hb2   = (const float*)d_in[14];
  float* outp = (float*)d_out;

  float* xbuf = (float*)d_ws;
  float* hbuf = xbuf + (size_t)N_NODES * ND;
  float* sums = hbuf + (size_t)N_NODES * ND;
  float* cnt  = sums + (size_t)N_GRAPH * HD;

  const int elemBlocks = (N_NODES * ND + 255) / 256;
  const int nNodeTiles = N_NODES / 16;  // 6250 exactly
  const int mlpBlocks = (nNodeTiles + MLP_WAVES - 1) / MLP_WAVES;

  for (int l = 0; l < NLAYERS; ++l) {
    const float* xin = (l == 0) ? x0 : xbuf;
    init_h_kernel<<<elemBlocks, 256, 0, stream>>>(xin, eps, l, hbuf);
    edge_kernel<<<1024, 32 * EDGE_WAVES, 0, stream>>>(xin, ei, ea,
                                                      linE + l * ED * ND, hbuf);
    node_mlp_kernel<<<mlpBlocks, 32 * MLP_WAVES, 0, stream>>>(
        hbuf, W1 + l * HD * HD, b1 + l * HD, W2 + l * HD * HD, b2 + l * HD,
        xbuf, nNodeTiles);
  }
  zero_pool_kernel<<<(N_GRAPH * HD + 255) / 256, 256, 0, stream>>>(sums, cnt);
  pool_kernel<<<elemBlocks, 256, 0, stream>>>(xbuf, batch, sums, cnt);
  head_kernel<<<(N_GRAPH + 255) / 256, 256, 0, stream>>>(sums, cnt, tc, hW1,
                                                         hb1, hW2, hb2, outp);
}